// GNN_30769145708770
// MI455X (gfx1250) — compile-verified
//
#include <hip/hip_runtime.h>

typedef float v2f __attribute__((ext_vector_type(2)));
typedef float v8f __attribute__((ext_vector_type(8)));

#define IN_DIM 128
#define HID 256

// ---------------- small utility kernels ----------------

__global__ void k_zero(int* __restrict__ p, int n) {
  int i = blockIdx.x * blockDim.x + threadIdx.x;
  if (i < n) p[i] = 0;
}

__global__ void k_count(const int* __restrict__ dst, int* __restrict__ cnt, int E) {
  int e = blockIdx.x * blockDim.x + threadIdx.x;
  if (e < E) atomicAdd(&cnt[dst[e]], 1);
}

// per-1024-block exclusive scan of cnt -> rowstart, block totals -> blockSums
__global__ void k_scan_block(const int* __restrict__ cnt, int* __restrict__ rowstart,
                             int* __restrict__ blockSums, int n) {
  __shared__ int sh[1024];
  int tid = threadIdx.x;
  int i = blockIdx.x * 1024 + tid;
  int v = (i < n) ? cnt[i] : 0;
  sh[tid] = v;
  __syncthreads();
  for (int off = 1; off < 1024; off <<= 1) {
    int t = (tid >= off) ? sh[tid - off] : 0;
    __syncthreads();
    sh[tid] += t;
    __syncthreads();
  }
  if (i < n) rowstart[i] = sh[tid] - v;  // exclusive
  if (tid == 1023) blockSums[blockIdx.x] = sh[tid];
}

// single-block exclusive scan of block sums (nb <= 1024)
__global__ void k_scan_sums(int* __restrict__ blockSums, int nb) {
  __shared__ int sh[1024];
  int tid = threadIdx.x;
  int v = (tid < nb) ? blockSums[tid] : 0;
  sh[tid] = v;
  __syncthreads();
  for (int off = 1; off < 1024; off <<= 1) {
    int t = (tid >= off) ? sh[tid - off] : 0;
    __syncthreads();
    sh[tid] += t;
    __syncthreads();
  }
  if (tid < nb) blockSums[tid] = sh[tid] - v;  // exclusive
}

__global__ void k_apply_off(int* __restrict__ rowstart, const int* __restrict__ blockSums,
                            int* __restrict__ cursor, int n) {
  int i = blockIdx.x * blockDim.x + threadIdx.x;
  if (i < n) {
    int r = rowstart[i] + blockSums[i >> 10];
    rowstart[i] = r;
    cursor[i] = r;
  }
}

__global__ void k_dinv(const int* __restrict__ cnt, float* __restrict__ dinv, int n) {
  int i = blockIdx.x * blockDim.x + threadIdx.x;
  if (i < n) dinv[i] = rsqrtf((float)(cnt[i] + 1));  // +1 = self loop; deg >= 1
}

__global__ void k_fill(const int* __restrict__ srcI, const int* __restrict__ dstI,
                       int* __restrict__ cursor, int* __restrict__ csr, int E) {
  int e = blockIdx.x * blockDim.x + threadIdx.x;
  if (e < E) {
    int p = atomicAdd(&cursor[dstI[e]], 1);
    csr[p] = srcI[e];
  }
}

// ---------------- WMMA GEMM: h = x @ W1 + b1 ----------------
// One wave computes a 16x64 output strip: 4 accumulators (16x16 each),
// K-loop of 32 steps with V_WMMA_F32_16X16X4_F32, A fragment reused 4x.
__global__ void k_gemm(const float* __restrict__ x, const float* __restrict__ W1,
                       const float* __restrict__ b1, float* __restrict__ h, int n) {
  int gtid = blockIdx.x * blockDim.x + threadIdx.x;
  int wave = gtid >> 5;
  int lane = threadIdx.x & 31;
  int nRowTiles = (n + 15) >> 4;
  int rowTile = wave >> 2;  // 4 column groups of 64
  int colGrp = wave & 3;
  if (rowTile >= nRowTiles) return;
  int row0 = rowTile << 4;
  int col0 = colGrp << 6;
  int laneM = lane & 15;  // row (A) / col (B,C) within tile
  int kh = lane >> 4;     // K half-select per ISA 16x4 A layout

  int arow = row0 + laneM;
  if (arow >= n) arow = n - 1;  // clamp (tail tiles); stores are guarded
  const float* A = x + (size_t)arow * IN_DIM;

  v8f c0 = {}, c1 = {}, c2 = {}, c3 = {};
  for (int kb = 0; kb < IN_DIM; kb += 4) {
    int k0 = kb + (kh << 1);
    v2f a;
    a.x = A[k0];
    a.y = A[k0 + 1];
    const float* B = W1 + (size_t)k0 * HID + col0 + laneM;
    v2f b;
    b.x = B[0];
    b.y = B[HID];
    c0 = __builtin_amdgcn_wmma_f32_16x16x4_f32(false, a, false, b, (short)0, c0, false, false);
    b.x = B[16];
    b.y = B[HID + 16];
    c1 = __builtin_amdgcn_wmma_f32_16x16x4_f32(false, a, false, b, (short)0, c1, false, false);
    b.x = B[32];
    b.y = B[HID + 32];
    c2 = __builtin_amdgcn_wmma_f32_16x16x4_f32(false, a, false, b, (short)0, c2, false, false);
    b.x = B[48];
    b.y = B[HID + 48];
    c3 = __builtin_amdgcn_wmma_f32_16x16x4_f32(false, a, false, b, (short)0, c3, false, false);
  }

  // C/D layout: VGPR r -> M = r (lanes 0-15) / M = r+8 (lanes 16-31), N = lane&15
  int rbase = kh * 8;
#pragma unroll
  for (int r = 0; r < 8; r++) {
    int row = row0 + rbase + r;
    if (row < n) {
      float* hp = h + (size_t)row * HID + col0 + laneM;
      hp[0]  = c0[r] + b1[col0 + laneM];
      hp[16] = c1[r] + b1[col0 + 16 + laneM];
      hp[32] = c2[r] + b1[col0 + 32 + laneM];
      hp[48] = c3[r] + b1[col0 + 48 + laneM];
    }
  }
}

// ---------------- fused layer-1 aggregate + bias + ReLU + (.)@W2 ----------------
// One wave32 per node; 8 channels/lane via two float4 loads (coalesced 1KB/edge).
__global__ void k_agg1(const float* __restrict__ h, const float* __restrict__ b1,
                       const float* __restrict__ W2, const float* __restrict__ dinv,
                       const int* __restrict__ rowstart, const int* __restrict__ cnt,
                       const int* __restrict__ csr, float* __restrict__ z, int n) {
  int wave = (blockIdx.x * blockDim.x + threadIdx.x) >> 5;
  int lane = threadIdx.x & 31;
  if (wave >= n) return;
  int i = wave;
  float di = dinv[i];
  float self = di * di;  // self-loop norm
  int cb = lane * 8;

  const float4* hv = (const float4*)(h + (size_t)i * HID + cb);
  float4 v0 = hv[0], v1 = hv[1];
  float acc[8] = {v0.x * self, v0.y * self, v0.z * self, v0.w * self,
                  v1.x * self, v1.y * self, v1.z * self, v1.w * self};

  int beg = rowstart[i], end = beg + cnt[i];
  for (int j = beg; j < end; j++) {
    int s = csr[j];
    float w = dinv[s] * di;
    const float4* sv = (const float4*)(h + (size_t)s * HID + cb);
    float4 u0 = sv[0], u1 = sv[1];
    acc[0] += u0.x * w; acc[1] += u0.y * w; acc[2] += u0.z * w; acc[3] += u0.w * w;
    acc[4] += u1.x * w; acc[5] += u1.y * w; acc[6] += u1.z * w; acc[7] += u1.w * w;
  }

  float p = 0.f;
#pragma unroll
  for (int t = 0; t < 8; t++) {
    float v = acc[t] + b1[cb + t];
    v = v > 0.f ? v : 0.f;  // ReLU
    p += v * W2[cb + t];    // layer-2 linear (HIDDEN -> 1)
  }
#pragma unroll
  for (int off = 16; off > 0; off >>= 1) p += __shfl_xor(p, off, 32);
  if (lane == 0) z[i] = p;
}

// ---------------- layer-2 aggregation (scalar per node) ----------------
__global__ void k_agg2(const float* __restrict__ z, const float* __restrict__ dinv,
                       const int* __restrict__ rowstart, const int* __restrict__ cnt,
                       const int* __restrict__ csr, const float* __restrict__ b2,
                       float* __restrict__ out, int n) {
  int i = blockIdx.x * blockDim.x + threadIdx.x;
  if (i >= n) return;
  float di = dinv[i];
  float s = z[i] * di * di;  // self loop
  int beg = rowstart[i], end = beg + cnt[i];
  for (int j = beg; j < end; j++) {
    int sI = csr[j];
    s += z[sI] * dinv[sI] * di;
  }
  out[i] = s + b2[0];
}

// ---------------- host launcher ----------------
extern "C" void kernel_launch(void* const* d_in, const int* in_sizes, int n_in,
                              void* d_out, int out_size, void* d_ws, size_t ws_size,
                              hipStream_t stream) {
  const float* x  = (const float*)d_in[0];
  const int*   ei = (const int*)d_in[1];   // edge_index [2, E]
  const float* W1 = (const float*)d_in[2];
  const float* b1 = (const float*)d_in[3];
  const float* W2 = (const float*)d_in[4];
  const float* b2 = (const float*)d_in[5];

  int n = in_sizes[0] / IN_DIM;
  int E = in_sizes[1] / 2;
  const int* srcI = ei;
  const int* dstI = ei + E;

  char* base = (char*)d_ws;
  size_t off = 0;
  auto alloc = [&](size_t bytes) -> void* {
    void* p = base + off;
    off = (off + bytes + 255) & ~(size_t)255;
    return p;
  };
  float* h        = (float*)alloc((size_t)n * HID * sizeof(float));  // 102.4 MB
  int*   cnt      = (int*)alloc((size_t)n * 4);
  int*   rowstart = (int*)alloc((size_t)n * 4);
  int*   cursor   = (int*)alloc((size_t)n * 4);
  int*   blockSums= (int*)alloc(4096);
  int*   csr      = (int*)alloc((size_t)E * 4);
  float* dinv     = (float*)alloc((size_t)n * 4);
  float* z        = (float*)alloc((size_t)n * 4);
  (void)ws_size; (void)n_in; (void)out_size;

  int nb256 = (n + 255) / 256;
  int eb256 = (E + 255) / 256;
  int nScan = (n + 1023) / 1024;

  k_zero<<<nb256, 256, 0, stream>>>(cnt, n);
  k_count<<<eb256, 256, 0, stream>>>(dstI, cnt, E);
  k_scan_block<<<nScan, 1024, 0, stream>>>(cnt, rowstart, blockSums, n);
  k_scan_sums<<<1, 1024, 0, stream>>>(blockSums, nScan);
  k_apply_off<<<nb256, 256, 0, stream>>>(rowstart, blockSums, cursor, n);
  k_dinv<<<nb256, 256, 0, stream>>>(cnt, dinv, n);
  k_fill<<<eb256, 256, 0, stream>>>(srcI, dstI, cursor, csr, E);

  int nRowTiles = (n + 15) / 16;
  int waves = nRowTiles * 4;          // 4 col groups of 64
  int gemmBlocks = (waves + 7) / 8;   // 8 waves per 256-thread block
  k_gemm<<<gemmBlocks, 256, 0, stream>>>(x, W1, b1, h, n);

  int agg1Blocks = (n + 7) / 8;       // one wave per node
  k_agg1<<<agg1Blocks, 256, 0, stream>>>(h, b1, W2, dinv, rowstart, cnt, csr, z, n);
  k_agg2<<<nb256, 256, 0, stream>>>(z, dinv, rowstart, cnt, csr, b2, (float*)d_out, n);
}